// WalkConv_69827578299061
// MI455X (gfx1250) — compile-verified
//
#include <hip/hip_runtime.h>

typedef float v2f __attribute__((ext_vector_type(2)));
typedef float v8f __attribute__((ext_vector_type(8)));

#define NN    512
#define HID   8
#define EDIM  4
#define NE    16384
#define KDIM  4096   // NN * HID
#define KB    32     // K-block staged into LDS per stage

// ---------------------------------------------------------------------------
// Pass 1: deterministic duplicate resolution — highest edge id wins a (j,k) slot
// ---------------------------------------------------------------------------
__global__ void winner_kernel(const int* __restrict__ ei, int* __restrict__ winner) {
    int e = blockIdx.x * blockDim.x + threadIdx.x;
    if (e >= NE) return;
    int e0 = ei[e];
    int e1 = ei[NE + e];
    atomicMax(&winner[e0 * NN + e1], e);
}

// ---------------------------------------------------------------------------
// Pass 2: per-edge MLP 4->32->32->32->64 (ReLU on first 3), scatter the 8x8
// block directly into the dense B matrix (4096 x 4096, row = j*8+c, col = k*8+t)
// ---------------------------------------------------------------------------
__global__ void edge_mlp_kernel(const int* __restrict__ ei, const float* __restrict__ ea,
                                const float* __restrict__ w0, const float* __restrict__ b0,
                                const float* __restrict__ w1, const float* __restrict__ b1,
                                const float* __restrict__ w2, const float* __restrict__ b2,
                                const float* __restrict__ w3, const float* __restrict__ b3,
                                const int* __restrict__ winner, float* __restrict__ Bmat) {
    int e = blockIdx.x * blockDim.x + threadIdx.x;
    if (e >= NE) return;

    float x[EDIM];
#pragma unroll
    for (int c = 0; c < EDIM; ++c) x[c] = ea[e * EDIM + c];

    float h0[32];
#pragma unroll
    for (int o = 0; o < 32; ++o) {
        float s = b0[o];
#pragma unroll
        for (int c = 0; c < EDIM; ++c) s = fmaf(x[c], w0[c * 32 + o], s);
        h0[o] = fmaxf(s, 0.0f);
    }
    float h1[32];
#pragma unroll
    for (int o = 0; o < 32; ++o) {
        float s = b1[o];
#pragma unroll
        for (int c = 0; c < 32; ++c) s = fmaf(h0[c], w1[c * 32 + o], s);
        h1[o] = fmaxf(s, 0.0f);
    }
    float h2[32];
#pragma unroll
    for (int o = 0; o < 32; ++o) {
        float s = b2[o];
#pragma unroll
        for (int c = 0; c < 32; ++c) s = fmaf(h1[c], w2[c * 32 + o], s);
        h2[o] = fmaxf(s, 0.0f);
    }

    int e0 = ei[e];
    int e1 = ei[NE + e];
    if (winner[e0 * NN + e1] != e) return;   // lost duplicate slot -> skip write

#pragma unroll
    for (int j = 0; j < HID * HID; ++j) {
        float s = b3[j];
#pragma unroll
        for (int c = 0; c < 32; ++c) s = fmaf(h2[c], w3[c * 64 + j], s);
        int crow = j >> 3;   // input channel c
        int tcol = j & 7;    // output channel t
        Bmat[(e0 * HID + crow) * KDIM + e1 * HID + tcol] = s;
    }
}

// ---------------------------------------------------------------------------
// Pass 3: mtr1 = A(512x4096) @ B(4096x4096) with V_WMMA_F32_16X16X4_F32.
// One block per 16-col stripe of C; 8 waves x 64 rows cover all 512 M rows,
// so the B fragment is identical across waves -> staged once into LDS by
// wave 0 with global_load_async_to_lds_b128 (double buffered, ASYNCcnt).
// Per K-step each wave: 4x b64 A loads + 1x ds_load_2addr b pair + 4 WMMAs.
// Fuses diag-zero and /HID into the store.
// ---------------------------------------------------------------------------
__global__ __launch_bounds__(256) void wmma_gemm_kernel(const float* __restrict__ A,
                                                        const float* __restrict__ Bm,
                                                        float* __restrict__ C) {
    __shared__ float bsh[2][KB * 16];   // 2 x 2KB: B[kb..kb+32) x [n0..n0+16)

    const int tid   = threadIdx.x;
    const int lane  = tid & 31;
    const int wave  = tid >> 5;
    const int n0    = blockIdx.x * 16;     // C column tile
    const int m     = lane & 15;           // row-in-tile (A) / col-in-tile (B,D)
    const int kp    = (lane >> 4) << 1;    // K pair offset: 0 or 2
    const int rbase = wave * 64;           // this wave's 64-row chunk

    const float* arow0 = A + (rbase +  0 + m) * KDIM + kp;
    const float* arow1 = A + (rbase + 16 + m) * KDIM + kp;
    const float* arow2 = A + (rbase + 32 + m) * KDIM + kp;
    const float* arow3 = A + (rbase + 48 + m) * KDIM + kp;

    v8f acc0 = {0.f,0.f,0.f,0.f,0.f,0.f,0.f,0.f};
    v8f acc1 = acc0, acc2 = acc0, acc3 = acc0;

    // staging geometry (wave 0 only): 32 lanes x 16B = 8 rows of 64B per op
    const int srow = lane >> 2;          // 0..7 row within 8-row group
    const int scol = (lane & 3) << 2;    // 0,4,8,12 floats (16B chunks)

    // prologue: stage K-block 0 into buffer 0
    if (tid < 32) {
#pragma unroll
        for (int i = 0; i < 4; ++i) {
            const float* g = Bm + (i * 8 + srow) * KDIM + n0 + scol;
            unsigned l = (unsigned)(unsigned long long)&bsh[0][(i * 8 + srow) * 16 + scol];
            asm volatile("global_load_async_to_lds_b128 %0, %1, off"
                         :: "v"(l), "v"(g) : "memory");
        }
    }

    int buf = 0;
#pragma unroll 1
    for (int kb = 0; kb < KDIM; kb += KB, buf ^= 1) {
        if (tid < 32) {                       // wave-uniform: EXEC stays full
            int nkb = kb + KB;
            if (nkb < KDIM) {
#pragma unroll
                for (int i = 0; i < 4; ++i) {
                    const float* g = Bm + (nkb + i * 8 + srow) * KDIM + n0 + scol;
                    unsigned l = (unsigned)(unsigned long long)
                                 &bsh[buf ^ 1][(i * 8 + srow) * 16 + scol];
                    asm volatile("global_load_async_to_lds_b128 %0, %1, off"
                                 :: "v"(l), "v"(g) : "memory");
                }
                // 4 newer ops outstanding; in-order completion => current stage done
                asm volatile("s_wait_asynccnt 0x4" ::: "memory");
            } else {
                asm volatile("s_wait_asynccnt 0x0" ::: "memory");
            }
        }
        __syncthreads();                      // publish bsh[buf] to all waves

        const float* bb = &bsh[buf][0];
#pragma unroll
        for (int kk = 0; kk < KB; kk += 4) {
            v2f a0 = *(const v2f*)(arow0 + kb + kk);
            v2f a1 = *(const v2f*)(arow1 + kb + kk);
            v2f a2 = *(const v2f*)(arow2 + kb + kk);
            v2f a3 = *(const v2f*)(arow3 + kb + kk);
            v2f b;
            b.x = bb[(kk + kp) * 16 + m];          // merges to ds_load_2addr_b32
            b.y = bb[(kk + kp + 1) * 16 + m];
            acc0 = __builtin_amdgcn_wmma_f32_16x16x4_f32(false, a0, false, b,
                                                         (short)0, acc0, false, false);
            acc1 = __builtin_amdgcn_wmma_f32_16x16x4_f32(false, a1, false, b,
                                                         (short)0, acc1, false, false);
            acc2 = __builtin_amdgcn_wmma_f32_16x16x4_f32(false, a2, false, b,
                                                         (short)0, acc2, false, false);
            acc3 = __builtin_amdgcn_wmma_f32_16x16x4_f32(false, a3, false, b,
                                                         (short)0, acc3, false, false);
        }
        __syncthreads();                      // all waves done reading bsh[buf]
    }

    // store 4 C tiles; D layout: VGPR r -> row r + 8*(lane/16), col = lane%16
    const int col  = n0 + m;
    const int kcol = col >> 3;
    const int roff = (lane >> 4) << 3;
#pragma unroll
    for (int t = 0; t < 4; ++t) {
        v8f acc = (t == 0) ? acc0 : (t == 1) ? acc1 : (t == 2) ? acc2 : acc3;
        int rb = rbase + t * 16 + roff;
#pragma unroll
        for (int r = 0; r < 8; ++r) {
            int row = rb + r;
            float v = (row == kcol) ? 0.0f : acc[r] * (1.0f / HID);
            C[row * KDIM + col] = v;
        }
    }
}

// ---------------------------------------------------------------------------
// Pass 4: sym = mtr1 + (mtr1^T(0,1) @ lin_w + lin_b); out = mtr + MLP(sym)
// MLP: 8->32->32->32->8, ReLU on first 3. One thread per (i,k) position.
// ---------------------------------------------------------------------------
__global__ void final_kernel(const float* __restrict__ mtr, const float* __restrict__ mtr1,
                             const float* __restrict__ lw, const float* __restrict__ lb,
                             const float* __restrict__ w0, const float* __restrict__ b0,
                             const float* __restrict__ w1, const float* __restrict__ b1,
                             const float* __restrict__ w2, const float* __restrict__ b2,
                             const float* __restrict__ w3, const float* __restrict__ b3,
                             float* __restrict__ out) {
    int tid = blockIdx.x * blockDim.x + threadIdx.x;
    if (tid >= NN * NN) return;
    int i = tid >> 9;
    int k = tid & (NN - 1);

    const float* m1 = mtr1 + i * KDIM + k * HID;   // mtr1[i,k,:]
    const float* mT = mtr1 + k * KDIM + i * HID;   // mtr1[k,i,:]

    float tfwd[HID], ttr[HID], sym[HID];
#pragma unroll
    for (int t = 0; t < HID; ++t) { tfwd[t] = m1[t]; ttr[t] = mT[t]; }
#pragma unroll
    for (int t = 0; t < HID; ++t) {
        float s = lb[t];
#pragma unroll
        for (int c = 0; c < HID; ++c) s = fmaf(ttr[c], lw[c * HID + t], s);
        sym[t] = tfwd[t] + s;
    }

    float h0[32];
#pragma unroll
    for (int o = 0; o < 32; ++o) {
        float s = b0[o];
#pragma unroll
        for (int c = 0; c < HID; ++c) s = fmaf(sym[c], w0[c * 32 + o], s);
        h0[o] = fmaxf(s, 0.0f);
    }
    float h1[32];
#pragma unroll
    for (int o = 0; o < 32; ++o) {
        float s = b1[o];
#pragma unroll
        for (int c = 0; c < 32; ++c) s = fmaf(h0[c], w1[c * 32 + o], s);
        h1[o] = fmaxf(s, 0.0f);
    }
    float h2[32];
#pragma unroll
    for (int o = 0; o < 32; ++o) {
        float s = b2[o];
#pragma unroll
        for (int c = 0; c < 32; ++c) s = fmaf(h1[c], w2[c * 32 + o], s);
        h2[o] = fmaxf(s, 0.0f);
    }
#pragma unroll
    for (int t = 0; t < HID; ++t) {
        float s = b3[t];
#pragma unroll
        for (int c = 0; c < 32; ++c) s = fmaf(h2[c], w3[c * HID + t], s);
        out[tid * HID + t] = mtr[tid * HID + t] + s;
    }
}

// ---------------------------------------------------------------------------
extern "C" void kernel_launch(void* const* d_in, const int* in_sizes, int n_in,
                              void* d_out, int out_size, void* d_ws, size_t ws_size,
                              hipStream_t stream) {
    (void)in_sizes; (void)n_in; (void)out_size; (void)ws_size;

    // setup_inputs() pytree-flattened order:
    const float* mtr = (const float*)d_in[0];            // (512,512,8) f32
    const int*   ei  = (const int*)d_in[1];              // (2,16384) int32
    const float* ea  = (const float*)d_in[2];            // (16384,4) f32
    const float* ew0 = (const float*)d_in[3];   const float* eb0 = (const float*)d_in[4];
    const float* ew1 = (const float*)d_in[5];   const float* eb1 = (const float*)d_in[6];
    const float* ew2 = (const float*)d_in[7];   const float* eb2 = (const float*)d_in[8];
    const float* ew3 = (const float*)d_in[9];   const float* eb3 = (const float*)d_in[10];
    const float* mw0 = (const float*)d_in[11];  const float* mb0 = (const float*)d_in[12];
    const float* mw1 = (const float*)d_in[13];  const float* mb1 = (const float*)d_in[14];
    const float* mw2 = (const float*)d_in[15];  const float* mb2 = (const float*)d_in[16];
    const float* mw3 = (const float*)d_in[17];  const float* mb3 = (const float*)d_in[18];
    const float* lw  = (const float*)d_in[19];
    const float* lb  = (const float*)d_in[20];
    float* out = (float*)d_out;

    // workspace: B (64MB) | mtr1 (8MB) | winner (1MB)
    char*  ws     = (char*)d_ws;
    float* Bmat   = (float*)ws;
    float* mtr1   = (float*)(ws + (size_t)KDIM * KDIM * sizeof(float));
    int*   winner = (int*)(ws + (size_t)KDIM * KDIM * sizeof(float)
                              + (size_t)NN * KDIM * sizeof(float));

    hipMemsetAsync(Bmat, 0, (size_t)KDIM * KDIM * sizeof(float), stream);
    hipMemsetAsync(winner, 0xFF, (size_t)NN * NN * sizeof(int), stream);  // -1

    winner_kernel<<<NE / 256, 256, 0, stream>>>(ei, winner);
    edge_mlp_kernel<<<NE / 256, 256, 0, stream>>>(ei, ea, ew0, eb0, ew1, eb1,
                                                  ew2, eb2, ew3, eb3, winner, Bmat);

    wmma_gemm_kernel<<<KDIM / 16, 256, 0, stream>>>(mtr, Bmat, mtr1);

    final_kernel<<<(NN * NN) / 256, 256, 0, stream>>>(mtr, mtr1, lw, lb,
                                                      mw0, mb0, mw1, mb1,
                                                      mw2, mb2, mw3, mb3, out);
}